// KPooling_23416161698206
// MI455X (gfx1250) — compile-verified
//
#include <hip/hip_runtime.h>
#include <math.h>

// ---------------------------------------------------------------------------
// GraphSAGE (mean) x3 + TopK pooling x3 + summarize + linear head, for
// B=8 graphs, N=10000 nodes, E=1.28M edges, F=H=128, OUT=64.
// Dense per-node GEMMs use CDNA5 V_WMMA_F32_16X16X4_F32 (full f32 fidelity;
// the pipeline is memory-bound on the edge scatter, so f32 WMMA is free).
// ---------------------------------------------------------------------------

typedef float v2f __attribute__((ext_vector_type(2)));
typedef float v8f __attribute__((ext_vector_type(8)));

#define HDIM 128

// ------------------------------- utilities ---------------------------------

__global__ void fill_u32_kernel(unsigned int* __restrict__ p, unsigned int v, long long n) {
    long long t = (long long)blockIdx.x * blockDim.x + threadIdx.x;
    if (t < n) p[t] = v;
}

__device__ __forceinline__ unsigned int sortable_key(float s) {
    unsigned int b = __float_as_uint(s);
    return (b & 0x80000000u) ? ~b : (b | 0x80000000u);
}

// ------------------------- edge scatter (mean aggr) ------------------------
// One wave (32 lanes) per edge; lane l moves features [4l, 4l+4) via float4.
__global__ void scatter_kernel(const float* __restrict__ x,
                               const int* __restrict__ src,
                               const int* __restrict__ dst,
                               const int* __restrict__ valid,
                               float* __restrict__ agg,
                               float* __restrict__ deg, int E) {
    long long t = (long long)blockIdx.x * blockDim.x + threadIdx.x;
    int e = (int)(t >> 5);
    int lane = (int)(t & 31);
    if (e >= E) return;
    if (valid && !valid[e]) return;
    int s = src[e], d = dst[e];
    const float4 v = *reinterpret_cast<const float4*>(x + (size_t)s * HDIM + lane * 4);
    float* ap = agg + (size_t)d * HDIM + lane * 4;
    atomicAdd(ap + 0, v.x);
    atomicAdd(ap + 1, v.y);
    atomicAdd(ap + 2, v.z);
    atomicAdd(ap + 3, v.w);
    if (lane == 0) atomicAdd(deg + d, 1.0f);
}

// --------------------- SAGE GEMM via V_WMMA_F32_16X16X4 --------------------
// y[m][n] = sum_k (agg[m][k]/max(deg,1)) * Wl[n][k] + sum_k x[m][k]*Wr[n][k] + bl[n]
// Block = 256 threads = 8 waves; wave w owns N-tile [16w,16w+16); block owns a
// 16-row M strip. K=128 -> 32 K-steps x 2 weight matrices = 64 WMMAs/tile.
// A-layout (16x4 f32): lanes 0-15 hold K={0,1}, lanes 16-31 K={2,3} of row M=lane&15.
// B-layout mirrors A with M<->N. C/D: VGPR j = row (j + 8*(lane>=16)), col lane&15.
__global__ void sage_gemm_wmma(const float* __restrict__ agg,
                               const float* __restrict__ deg,
                               const float* __restrict__ xin,
                               const float* __restrict__ Wl,
                               const float* __restrict__ bl,
                               const float* __restrict__ Wr,
                               float* __restrict__ y,
                               int n_nodes, int do_relu) {
    const int lane = threadIdx.x & 31;
    const int wave = threadIdx.x >> 5;   // N tile id (0..7)
    const int row0 = blockIdx.x * 16;
    const int half = lane >> 4;          // 0 or 1
    const int l16  = lane & 15;
    const int m = row0 + l16;
    const int n = wave * 16 + l16;
    const int koff = half * 2;

    const float invd = 1.0f / fmaxf(deg[m], 1.0f);
    const float* __restrict__ arow  = agg + (size_t)m * HDIM;
    const float* __restrict__ xrow  = xin + (size_t)m * HDIM;
    const float* __restrict__ wlrow = Wl  + (size_t)n * HDIM;
    const float* __restrict__ wrrow = Wr  + (size_t)n * HDIM;

    v8f acc = {0.f, 0.f, 0.f, 0.f, 0.f, 0.f, 0.f, 0.f};
#pragma unroll 8
    for (int kk = 0; kk < HDIM; kk += 4) {
        v2f a1, b1, a2, b2;
        a1.x = arow[kk + koff] * invd;
        a1.y = arow[kk + koff + 1] * invd;
        b1.x = wlrow[kk + koff];
        b1.y = wlrow[kk + koff + 1];
        acc = __builtin_amdgcn_wmma_f32_16x16x4_f32(false, a1, false, b1,
                                                    (short)0, acc, false, false);
        a2.x = xrow[kk + koff];
        a2.y = xrow[kk + koff + 1];
        b2.x = wrrow[kk + koff];
        b2.y = wrrow[kk + koff + 1];
        acc = __builtin_amdgcn_wmma_f32_16x16x4_f32(false, a2, false, b2,
                                                    (short)0, acc, false, false);
    }

    const float bias = bl[n];
#pragma unroll
    for (int j = 0; j < 8; ++j) {
        float v = acc[j] + bias;
        if (do_relu) v = fmaxf(v, 0.f);
        const int row = row0 + j + half * 8;
        y[(size_t)row * HDIM + n] = v;
    }
}

// ------------------------------ pooling score ------------------------------

__global__ void wnorm_kernel(const float* __restrict__ w, float* __restrict__ inv_norm) {
    int lane = threadIdx.x;   // single wave of 32
    float s = 0.f;
    for (int f = lane; f < HDIM; f += 32) { float v = w[f]; s += v * v; }
    for (int off = 16; off > 0; off >>= 1) s += __shfl_xor(s, off, 32);
    if (lane == 0) *inv_norm = 1.0f / sqrtf(s);
}

__global__ void score_kernel(const float* __restrict__ x, const float* __restrict__ w,
                             const float* __restrict__ inv_norm,
                             float* __restrict__ sc, int n) {
    long long t = (long long)blockIdx.x * blockDim.x + threadIdx.x;
    int node = (int)(t >> 5), lane = (int)(t & 31);
    if (node >= n) return;
    const float* xr = x + (size_t)node * HDIM;
    float s = xr[lane]      * w[lane]      + xr[lane + 32] * w[lane + 32] +
              xr[lane + 64] * w[lane + 64] + xr[lane + 96] * w[lane + 96];
    for (int off = 16; off > 0; off >>= 1) s += __shfl_xor(s, off, 32);
    if (lane == 0) sc[node] = tanhf(s * (*inv_norm));
}

// ------------------- exact per-graph top-k (radix select) ------------------
// One block per graph; 4 MSB-first passes over 256-bin LDS histograms.
__global__ void topk_threshold_kernel(const float* __restrict__ sc, int n_per, int k,
                                      unsigned int* __restrict__ thr,
                                      int* __restrict__ tieneed) {
    __shared__ unsigned int hist[256];
    __shared__ unsigned int sh_prefix;
    __shared__ int sh_need;
    const int b = blockIdx.x;
    const int tid = threadIdx.x;
    const float* base = sc + (size_t)b * n_per;
    if (tid == 0) { sh_prefix = 0u; sh_need = k; }
    __syncthreads();
    for (int d = 3; d >= 0; --d) {
        const int shift = d * 8;
        hist[tid] = 0u;
        __syncthreads();
        const unsigned int prefix = sh_prefix;
        const unsigned int himask = (d == 3) ? 0u : (0xFFFFFFFFu << (shift + 8));
        for (int i = tid; i < n_per; i += 256) {
            unsigned int u = sortable_key(base[i]);
            if ((u & himask) == prefix)
                atomicAdd(&hist[(u >> shift) & 255u], 1u);
        }
        __syncthreads();
        if (tid == 0) {
            int need = sh_need;
            unsigned int acc = 0;
            int chosen = 0;
            for (int bin = 255; bin >= 0; --bin) {
                unsigned int c = hist[bin];
                if ((int)(acc + c) >= need) { chosen = bin; break; }
                acc += c;
            }
            sh_need = need - (int)acc;
            sh_prefix = prefix | ((unsigned int)chosen << shift);
        }
        __syncthreads();
    }
    if (tid == 0) { thr[b] = sh_prefix; tieneed[b] = sh_need; }
}

// Keep all u > T plus exactly tieneed[b] ties; write perm/vals/node_map.
__global__ void compact_kernel(const float* __restrict__ sc,
                               const unsigned int* __restrict__ thr,
                               const int* __restrict__ tieneed,
                               int* __restrict__ cnt_keep, int* __restrict__ cnt_tie,
                               int n_per, int k,
                               int* __restrict__ perm, float* __restrict__ vals,
                               int* __restrict__ node_map, int total) {
    int i = blockIdx.x * blockDim.x + threadIdx.x;
    if (i >= total) return;
    const int b = i / n_per;
    const float s = sc[i];
    const unsigned int u = sortable_key(s);
    const unsigned int T = thr[b];
    int pos = -1;
    if (u > T) {
        pos = atomicAdd(&cnt_keep[b], 1);
    } else if (u == T) {
        int t = atomicAdd(&cnt_tie[b], 1);
        if (t < tieneed[b]) pos = atomicAdd(&cnt_keep[b], 1);
    }
    if (pos >= 0) {
        const int nid = b * k + pos;
        perm[nid] = i;
        vals[nid] = s;
        node_map[i] = nid;
    }
}

// x_new[nid] = x_old[perm[nid]] * vals[nid]; wave per node, float4 per lane.
__global__ void gather_kernel(const float* __restrict__ xin, const int* __restrict__ perm,
                              const float* __restrict__ vals, float* __restrict__ xout,
                              int n_new) {
    long long t = (long long)blockIdx.x * blockDim.x + threadIdx.x;
    int nid = (int)(t >> 5), lane = (int)(t & 31);
    if (nid >= n_new) return;
    const int old = perm[nid];
    const float sv = vals[nid];
    float4 v = *reinterpret_cast<const float4*>(xin + (size_t)old * HDIM + lane * 4);
    v.x *= sv; v.y *= sv; v.z *= sv; v.w *= sv;
    *reinterpret_cast<float4*>(xout + (size_t)nid * HDIM + lane * 4) = v;
}

__global__ void remap_kernel(const int* __restrict__ src_in, const int* __restrict__ dst_in,
                             const int* __restrict__ valid_in, const int* __restrict__ node_map,
                             int* __restrict__ src_out, int* __restrict__ dst_out,
                             int* __restrict__ valid_out, int E) {
    int e = blockIdx.x * blockDim.x + threadIdx.x;
    if (e >= E) return;
    const int ns = node_map[src_in[e]];
    const int nd = node_map[dst_in[e]];
    int v = (ns >= 0) && (nd >= 0);
    if (valid_in) v = v && valid_in[e];
    src_out[e] = v ? ns : 0;
    dst_out[e] = v ? nd : 0;
    valid_out[e] = v;
}

// summary[b][0:128] += mean over k rows; summary[b][128:256] += max over k rows.
__global__ void summarize_kernel(const float* __restrict__ x, int k,
                                 float* __restrict__ summary) {
    int t = blockIdx.x * blockDim.x + threadIdx.x;
    if (t >= 8 * HDIM) return;
    const int b = t >> 7, h = t & 127;
    float sum = 0.f, mx = -INFINITY;
    const float* p = x + (size_t)b * k * HDIM + h;
    for (int i = 0; i < k; ++i) {
        float v = p[(size_t)i * HDIM];
        sum += v;
        mx = fmaxf(mx, v);
    }
    summary[b * 256 + h]       += sum / (float)k;
    summary[b * 256 + 128 + h] += mx;
}

__global__ void out_kernel(const float* __restrict__ summary, const float* __restrict__ W,
                           const float* __restrict__ bias, float* __restrict__ out) {
    int t = blockIdx.x * blockDim.x + threadIdx.x;
    if (t >= 8 * 64) return;
    const int b = t >> 6, o = t & 63;
    float s = bias[o];
    const float* sr = summary + b * 256;
    const float* wr = W + o * 256;
    for (int j = 0; j < 256; ++j) s += sr[j] * wr[j];
    out[t] = s;
}

// ------------------------------- host driver -------------------------------

static inline unsigned cdiv(long long a, long long b) { return (unsigned)((a + b - 1) / b); }

extern "C" void kernel_launch(void* const* d_in, const int* in_sizes, int n_in,
                              void* d_out, int out_size, void* d_ws, size_t ws_size,
                              hipStream_t stream) {
    const float* x0      = (const float*)d_in[0];
    const int*   eidx    = (const int*)d_in[1];
    const float* w1l_W   = (const float*)d_in[2];
    const float* w1l_b   = (const float*)d_in[3];
    const float* w1r_W   = (const float*)d_in[4];
    const float* pool1_w = (const float*)d_in[5];
    const float* w2l_W   = (const float*)d_in[6];
    const float* w2l_b   = (const float*)d_in[7];
    const float* w2r_W   = (const float*)d_in[8];
    const float* pool2_w = (const float*)d_in[9];
    const float* w3l_W   = (const float*)d_in[10];
    const float* w3l_b   = (const float*)d_in[11];
    const float* w3r_W   = (const float*)d_in[12];
    const float* pool3_w = (const float*)d_in[13];
    const float* out_W   = (const float*)d_in[14];
    const float* out_b   = (const float*)d_in[15];
    float* out = (float*)d_out;

    const int Bg = 8;
    const int n0 = in_sizes[0] / HDIM;        // 80000
    const int Np = n0 / Bg;                   // 10000 nodes per graph
    const int E  = in_sizes[1] / 2;           // 1,280,000
    const int K1 = (Np + 1) / 2;              // 5000
    const int K2 = (K1 + 1) / 2;              // 2500
    const int K3 = (K2 + 1) / 2;              // 1250
    const int n1 = Bg * K1, n2 = Bg * K2, n3 = Bg * K3;
    const int* src0 = eidx;
    const int* dst0 = eidx + E;

    // ---- carve workspace (all buffers written before read) ----
    char* wp = (char*)d_ws;
    auto carve = [&](size_t bytes) -> void* {
        void* p = (void*)wp;
        wp += (bytes + 255) & ~(size_t)255;
        return p;
    };
    float* agg      = (float*)carve((size_t)n0 * HDIM * 4);
    float* bufA     = (float*)carve((size_t)n0 * HDIM * 4);
    float* bufB     = (float*)carve((size_t)n0 * HDIM * 4);
    float* deg      = (float*)carve((size_t)n0 * 4);
    float* sc       = (float*)carve((size_t)n0 * 4);
    int*   node_map = (int*)carve((size_t)n0 * 4);
    int*   perm     = (int*)carve((size_t)n1 * 4);
    float* vals     = (float*)carve((size_t)n1 * 4);
    int*   srcW     = (int*)carve((size_t)E * 4);
    int*   dstW     = (int*)carve((size_t)E * 4);
    int*   valW     = (int*)carve((size_t)E * 4);
    unsigned int* thr = (unsigned int*)carve(Bg * 4);
    int*   tieneed  = (int*)carve(Bg * 4);
    int*   cnt_keep = (int*)carve(Bg * 4);
    int*   cnt_tie  = (int*)carve(Bg * 4);
    float* inv_norm = (float*)carve(4);
    float* summary  = (float*)carve(Bg * 256 * 4);

    const int TB = 256;
    fill_u32_kernel<<<cdiv(Bg * 256, TB), TB, 0, stream>>>((unsigned int*)summary, 0u, Bg * 256);

    auto zero_agg = [&](int n) {
        fill_u32_kernel<<<cdiv((long long)n * HDIM, TB), TB, 0, stream>>>((unsigned int*)agg, 0u, (long long)n * HDIM);
        fill_u32_kernel<<<cdiv(n, TB), TB, 0, stream>>>((unsigned int*)deg, 0u, n);
    };
    auto pool = [&](const float* xin, const float* pw, int n_nodes, int n_per, int k,
                    float* xout) {
        wnorm_kernel<<<1, 32, 0, stream>>>(pw, inv_norm);
        score_kernel<<<cdiv((long long)n_nodes * 32, TB), TB, 0, stream>>>(xin, pw, inv_norm, sc, n_nodes);
        fill_u32_kernel<<<cdiv(n_nodes, TB), TB, 0, stream>>>((unsigned int*)node_map, 0xFFFFFFFFu, n_nodes);
        fill_u32_kernel<<<1, TB, 0, stream>>>((unsigned int*)cnt_keep, 0u, Bg);
        fill_u32_kernel<<<1, TB, 0, stream>>>((unsigned int*)cnt_tie, 0u, Bg);
        topk_threshold_kernel<<<Bg, 256, 0, stream>>>(sc, n_per, k, thr, tieneed);
        compact_kernel<<<cdiv(n_nodes, TB), TB, 0, stream>>>(sc, thr, tieneed, cnt_keep, cnt_tie,
                                                             n_per, k, perm, vals, node_map, n_nodes);
        gather_kernel<<<cdiv((long long)(Bg * k) * 32, TB), TB, 0, stream>>>(xin, perm, vals, xout, Bg * k);
    };

    // ================= layer 1 =================
    zero_agg(n0);
    scatter_kernel<<<cdiv((long long)E * 32, TB), TB, 0, stream>>>(x0, src0, dst0, nullptr, agg, deg, E);
    sage_gemm_wmma<<<n0 / 16, 256, 0, stream>>>(agg, deg, x0, w1l_W, w1l_b, w1r_W, bufA, n0, 0);
    pool(bufA, pool1_w, n0, Np, K1, bufB);
    remap_kernel<<<cdiv(E, TB), TB, 0, stream>>>(src0, dst0, nullptr, node_map, srcW, dstW, valW, E);
    summarize_kernel<<<cdiv(Bg * HDIM, TB), TB, 0, stream>>>(bufB, K1, summary);

    // ================= layer 2 =================
    zero_agg(n1);
    scatter_kernel<<<cdiv((long long)E * 32, TB), TB, 0, stream>>>(bufB, srcW, dstW, valW, agg, deg, E);
    sage_gemm_wmma<<<n1 / 16, 256, 0, stream>>>(agg, deg, bufB, w2l_W, w2l_b, w2r_W, bufA, n1, 1);
    pool(bufA, pool2_w, n1, K1, K2, bufB);
    remap_kernel<<<cdiv(E, TB), TB, 0, stream>>>(srcW, dstW, valW, node_map, srcW, dstW, valW, E);
    summarize_kernel<<<cdiv(Bg * HDIM, TB), TB, 0, stream>>>(bufB, K2, summary);

    // ================= layer 3 =================
    zero_agg(n2);
    scatter_kernel<<<cdiv((long long)E * 32, TB), TB, 0, stream>>>(bufB, srcW, dstW, valW, agg, deg, E);
    sage_gemm_wmma<<<n2 / 16, 256, 0, stream>>>(agg, deg, bufB, w3l_W, w3l_b, w3r_W, bufA, n2, 0);
    pool(bufA, pool3_w, n2, K2, K3, bufB);
    summarize_kernel<<<cdiv(Bg * HDIM, TB), TB, 0, stream>>>(bufB, K3, summary);

    // ================= head =================
    out_kernel<<<cdiv(Bg * 64, TB), TB, 0, stream>>>(summary, out_W, out_b, out);
    (void)n3; (void)ws_size; (void)n_in; (void)out_size;
}